// NodeAggregationConv_56023553409778
// MI455X (gfx1250) — compile-verified
//
#include <hip/hip_runtime.h>
#include <hip/hip_bf16.h>

// ---------------------------------------------------------------------------
// GIN (3 layers, eps=0) + JumpingKnowledge(cat) + Linear + ReLU
//   per layer: agg = scatter_add(h[src] -> dst); m = h + agg
//              h = relu( relu(m@W1+b1) @ W2 + b2 )
//   out = relu( cat(h1,h2,h3) @ lin_W + lin_b )
//
// Bandwidth-bound on the edge scatter (~2.5 GB @ 23.3 TB/s); GEMMs done with
// V_WMMA_F32_16X16X4_F32 (exact fp32, matches reference numerics).
// ---------------------------------------------------------------------------

#define N_NODES 100000
#define N_EDGES 1600000
#define HID 64
#define WAVES_PER_BLOCK 5   // 6250 tiles of 16 rows = 1250 blocks * 5 waves, exact

typedef __attribute__((ext_vector_type(2))) float v2f;
typedef __attribute__((ext_vector_type(8))) float v8f;

// ---------------------------------------------------------------------------
// Zero fill
// ---------------------------------------------------------------------------
__global__ void zero_f32(float* __restrict__ p, int n) {
    int i = blockIdx.x * blockDim.x + threadIdx.x;
    if (i < n) p[i] = 0.0f;
}

// ---------------------------------------------------------------------------
// Scatter-add: agg[dst[e], :] += h[src[e], :]   (16 threads per edge, float4)
// ---------------------------------------------------------------------------
__global__ void scatter_add(const float* __restrict__ h, int hstride,
                            const int* __restrict__ src,
                            const int* __restrict__ dst,
                            float* __restrict__ agg) {
    long long idx = (long long)blockIdx.x * blockDim.x + threadIdx.x;
    int e  = (int)(idx >> 4);
    int f4 = (int)(idx & 15) << 2;
    if (e >= N_EDGES) return;
    int s = src[e];
    int d = dst[e];
    const float4 v = *(const float4*)(h + (size_t)s * hstride + f4);
    float* a = agg + (size_t)d * HID + f4;
    atomicAdd(a + 0, v.x);
    atomicAdd(a + 1, v.y);
    atomicAdd(a + 2, v.z);
    atomicAdd(a + 3, v.w);
}

// ---------------------------------------------------------------------------
// One 16x64 @ 64x64 accumulate using V_WMMA_F32_16X16X4_F32.
// A fragment (16x4 f32): lanes 0-15 hold M=lane, lanes 16-31 hold M=lane-16;
//   VGPR v holds K = (lane>>4)*2 + v within the group of 4.
// B fragment (4x16 f32): lane holds N=lane&15; VGPR v holds K=(lane>>4)*2+v.
// C/D (16x16 f32): lane holds N=lane&15; VGPR v holds M = v + 8*(lane>>4).
// ---------------------------------------------------------------------------
__device__ __forceinline__ void gemm_16x64_64x64(const float* __restrict__ tileA, // LDS 16x64
                                                 const float* __restrict__ sW,    // LDS 64x64
                                                 int lane, v8f acc[4]) {
    const int row = lane & 15;
    const int kb  = (lane >> 4) * 2;
    const int col = lane & 15;
#pragma unroll
    for (int k = 0; k < 16; ++k) {
        v2f a;
        a.x = tileA[row * HID + k * 4 + kb + 0];
        a.y = tileA[row * HID + k * 4 + kb + 1];
#pragma unroll
        for (int n = 0; n < 4; ++n) {
            v2f b;
            b.x = sW[(k * 4 + kb + 0) * HID + n * 16 + col];
            b.y = sW[(k * 4 + kb + 1) * HID + n * 16 + col];
            acc[n] = __builtin_amdgcn_wmma_f32_16x16x4_f32(
                false, a, false, b, (short)0, acc[n], false, false);
        }
    }
}

// Stage 16 rows x 64 cols of (hin [+ agg]) into a wave-private LDS tile.
__device__ __forceinline__ void stage_tile(const float* __restrict__ hin, int hstride,
                                           const float* __restrict__ agg, // may be null
                                           size_t rowBase, int lane,
                                           float* __restrict__ myTile) {
    const int r  = lane >> 1;
    const int c0 = (lane & 1) * 32;
    const float* hrow = hin + (rowBase + r) * (size_t)hstride + c0;
    float* trow = myTile + r * HID + c0;
    if (agg) {
        const float* arow = agg + (rowBase + r) * (size_t)HID + c0;
#pragma unroll
        for (int c = 0; c < 32; c += 4) {
            float4 hv = *(const float4*)(hrow + c);
            float4 av = *(const float4*)(arow + c);
            float4 m  = make_float4(hv.x + av.x, hv.y + av.y, hv.z + av.z, hv.w + av.w);
            *(float4*)(trow + c) = m;
        }
    } else {
#pragma unroll
        for (int c = 0; c < 32; c += 4) {
            *(float4*)(trow + c) = *(const float4*)(hrow + c);
        }
    }
}

// ---------------------------------------------------------------------------
// GIN layer MLP: out[:, :] = relu( relu((hin+agg)@W1+b1) @ W2 + b2 )
// out written with row stride `ostride` (into the JK concat buffer).
// ---------------------------------------------------------------------------
__global__ __launch_bounds__(32 * WAVES_PER_BLOCK)
void gin_mlp_wmma(const float* __restrict__ hin, int hstride,
                  const float* __restrict__ agg,
                  const float* __restrict__ W1, const float* __restrict__ b1,
                  const float* __restrict__ W2, const float* __restrict__ b2,
                  float* __restrict__ out, int ostride) {
    __shared__ float sW1[HID * HID];
    __shared__ float sW2[HID * HID];
    __shared__ float sb1[HID];
    __shared__ float sb2[HID];
    __shared__ float tiles[WAVES_PER_BLOCK][16 * HID];

    const int t = threadIdx.x;
    for (int i = t; i < HID * HID; i += 32 * WAVES_PER_BLOCK) {
        sW1[i] = W1[i];
        sW2[i] = W2[i];
    }
    if (t < HID) { sb1[t] = b1[t]; sb2[t] = b2[t]; }
    __syncthreads();

    const int wave = t >> 5;
    const int lane = t & 31;
    const int tileId = blockIdx.x * WAVES_PER_BLOCK + wave; // always valid
    const size_t rowBase = (size_t)tileId * 16;
    float* myTile = tiles[wave];

    // Stage A = hin + agg
    stage_tile(hin, hstride, agg, rowBase, lane, myTile);
    __syncthreads();

    // GEMM1
    v8f acc[4] = {};
    gemm_16x64_64x64(myTile, sW1, lane, acc);

    // bias + relu -> back to LDS tile (re-fragment for GEMM2)
    const int col = lane & 15;
    const int mhalf = (lane >> 4) * 8;
#pragma unroll
    for (int n = 0; n < 4; ++n) {
        const float bv = sb1[n * 16 + col];
#pragma unroll
        for (int v = 0; v < 8; ++v) {
            float x = acc[n][v] + bv;
            myTile[(mhalf + v) * HID + n * 16 + col] = fmaxf(x, 0.0f);
        }
    }
    __syncthreads();

    // GEMM2
    v8f acc2[4] = {};
    gemm_16x64_64x64(myTile, sW2, lane, acc2);

    // bias + relu -> global (JK concat slot)
#pragma unroll
    for (int n = 0; n < 4; ++n) {
        const float bv = sb2[n * 16 + col];
#pragma unroll
        for (int v = 0; v < 8; ++v) {
            float x = acc2[n][v] + bv;
            out[(rowBase + mhalf + v) * (size_t)ostride + n * 16 + col] = fmaxf(x, 0.0f);
        }
    }
}

// ---------------------------------------------------------------------------
// Final: out = relu( xs(100000x192) @ lin_W(192x64) + lin_b )
// K=192 processed as 3 chunks of 64 with a shared 64x64 LDS weight buffer.
// ---------------------------------------------------------------------------
__global__ __launch_bounds__(32 * WAVES_PER_BLOCK)
void final_linear_wmma(const float* __restrict__ xs,   // stride 192
                       const float* __restrict__ linW, // 192 x 64 row-major
                       const float* __restrict__ linb,
                       float* __restrict__ out) {      // stride 64
    __shared__ float sW[HID * HID];
    __shared__ float sb[HID];
    __shared__ float tiles[WAVES_PER_BLOCK][16 * HID];

    const int t = threadIdx.x;
    if (t < HID) sb[t] = linb[t];

    const int wave = t >> 5;
    const int lane = t & 31;
    const int tileId = blockIdx.x * WAVES_PER_BLOCK + wave; // always valid
    const size_t rowBase = (size_t)tileId * 16;
    float* myTile = tiles[wave];

    v8f acc[4] = {};
#pragma unroll
    for (int c = 0; c < 3; ++c) {
        __syncthreads(); // protect sW from previous chunk's readers
        for (int i = t; i < HID * HID; i += 32 * WAVES_PER_BLOCK)
            sW[i] = linW[c * HID * HID + i];
        stage_tile(xs + c * HID, 3 * HID, nullptr, rowBase, lane, myTile);
        __syncthreads();
        gemm_16x64_64x64(myTile, sW, lane, acc);
    }

    const int col = lane & 15;
    const int mhalf = (lane >> 4) * 8;
#pragma unroll
    for (int n = 0; n < 4; ++n) {
        const float bv = sb[n * 16 + col];
#pragma unroll
        for (int v = 0; v < 8; ++v) {
            float x = acc[n][v] + bv;
            out[(rowBase + mhalf + v) * (size_t)HID + n * 16 + col] = fmaxf(x, 0.0f);
        }
    }
}

// ---------------------------------------------------------------------------
// Launch
// ---------------------------------------------------------------------------
extern "C" void kernel_launch(void* const* d_in, const int* in_sizes, int n_in,
                              void* d_out, int out_size, void* d_ws, size_t ws_size,
                              hipStream_t stream) {
    (void)in_sizes; (void)n_in; (void)out_size; (void)ws_size;

    const float* x   = (const float*)d_in[0];
    const int*   ei  = (const int*)d_in[1];
    const int*   src = ei;
    const int*   dst = ei + N_EDGES;
    const float* linW = (const float*)d_in[14];
    const float* linb = (const float*)d_in[15];

    float* agg = (float*)d_ws;                 // N_NODES * 64
    float* xs  = agg + (size_t)N_NODES * HID;  // N_NODES * 192 (JK concat)

    const int nTiles  = N_NODES / 16;                  // 6250
    const int mlpGrid = nTiles / WAVES_PER_BLOCK;      // 1250
    const int zeroGrid = (N_NODES * HID + 255) / 256;
    const long long scatterThreads = (long long)N_EDGES * 16;
    const int scatterGrid = (int)((scatterThreads + 255) / 256);

    const float* hin = x;
    int hstride = HID;
    for (int l = 0; l < 3; ++l) {
        const float* W1 = (const float*)d_in[2 + 4 * l + 0];
        const float* b1 = (const float*)d_in[2 + 4 * l + 1];
        const float* W2 = (const float*)d_in[2 + 4 * l + 2];
        const float* b2 = (const float*)d_in[2 + 4 * l + 3];

        zero_f32<<<zeroGrid, 256, 0, stream>>>(agg, N_NODES * HID);
        scatter_add<<<scatterGrid, 256, 0, stream>>>(hin, hstride, src, dst, agg);
        gin_mlp_wmma<<<mlpGrid, 32 * WAVES_PER_BLOCK, 0, stream>>>(
            hin, hstride, agg, W1, b1, W2, b2, xs + l * HID, 3 * HID);

        hin = xs + l * HID;
        hstride = 3 * HID;
    }

    final_linear_wmma<<<mlpGrid, 32 * WAVES_PER_BLOCK, 0, stream>>>(
        xs, linW, linb, (float*)d_out);
}